// NUSpectralConv1D_44633300140642
// MI455X (gfx1250) — compile-verified
//
#include <hip/hip_runtime.h>
#include <hip/hip_bf16.h>
#include <math.h>

#define BN 8
#define CC 128
#define NN 16384
#define MM 257
#define KK 128          // max |mode|
#define TE 256
#define JP 272          // padded interleaved basis dim: 2*(KK+1)=258 -> 272
#define JT 17           // JP/16 tiles

typedef float v2f __attribute__((ext_vector_type(2)));
typedef float v8f __attribute__((ext_vector_type(8)));

// ---------------------------------------------------------------------------
// Kernel 1: shift[b][c] = tproj_b[c] + sum_t silu(temb[b][t]) * tproj_w[c][t]
// ---------------------------------------------------------------------------
__global__ __launch_bounds__(CC)
void shift_kernel(const float* __restrict__ temb, const float* __restrict__ w,
                  const float* __restrict__ bvec, float* __restrict__ shift) {
    int b = blockIdx.x, c = threadIdx.x;
    const float* tr = temb + b * TE;
    const float* wc = w + c * TE;
    float s = 0.f;
    for (int t = 0; t < TE; ++t) {
        float x  = tr[t];
        float sl = x / (1.f + __expf(-x));   // SiLU
        s += sl * wc[t];
    }
    shift[b * CC + c] = s + bvec[c];
}

// ---------------------------------------------------------------------------
// Kernel 2 (forward NUFFT GEMM): P[b][c][2k]   = sum_n (x+shift) cos(k z_n)
//                                P[b][c][2k+1] = sum_n (x+shift) sin(k z_n)
// One wave per block; one 16(c) x 16(j) output tile; K-loop over N in 32-n
// chunks. Trig basis built per chunk via angle recurrence in LDS.
// grid = (ct=8, jt=17, b=8), block = 32
// ---------------------------------------------------------------------------
__global__ __launch_bounds__(32)
void fwd_kernel(const float* __restrict__ x_feat, const float* __restrict__ z,
                const float* __restrict__ shift, float* __restrict__ P) {
    const int ct = blockIdx.x, jt = blockIdx.y, b = blockIdx.z;
    const int lane = threadIdx.x;
    const int col  = lane & 15;      // A: M row (c); B: N col (j)
    const int hi   = lane >> 4;      // selects K pair {2*hi, 2*hi+1}

    __shared__ float xt[16 * 36];    // x tile [c][n], stride 36 (16B aligned, conflict-free)
    __shared__ float tg[32 * 18];    // trig   [n][j], stride 18 (conflict-free)

    const float* xb = x_feat + (size_t)b * CC * NN;
    const float* zb = z + (size_t)b * NN;
    const int    k0 = jt * 8;        // first k of this 16-col (8-mode) tile

    v8f acc = {};

    for (int n0 = 0; n0 < NN; n0 += 32) {
        __syncthreads();
        // ---- stage x tile: 16 rows x 32 cols, coalesced float4 loads ----
        #pragma unroll
        for (int p = 0; p < 4; ++p) {
            int r  = (lane >> 3) + p * 4;
            int q  = lane & 7;
            int cg = ct * 16 + r;
            float4 v = *(const float4*)(xb + (size_t)cg * NN + n0 + q * 4);
            float  s = shift[b * CC + cg];
            float* d = &xt[r * 36 + q * 4];
            d[0] = v.x + s; d[1] = v.y + s; d[2] = v.z + s; d[3] = v.w + s;
        }
        // ---- stage trig tile: thread t owns n = n0+t, modes k0..k0+7 ----
        {
            float zz = zb[n0 + lane];
            float c1, s1, ck, sk;
            sincosf(zz, &s1, &c1);
            sincosf((float)k0 * zz, &sk, &ck);
            #pragma unroll
            for (int i = 0; i < 8; ++i) {
                tg[lane * 18 + 2 * i]     = ck;
                tg[lane * 18 + 2 * i + 1] = sk;
                float cn = ck * c1 - sk * s1;   // angle recurrence
                sk = sk * c1 + ck * s1;
                ck = cn;
            }
        }
        __syncthreads();
        // ---- 8 WMMA steps (K=4 each) over the 32-n chunk ----
        #pragma unroll
        for (int s = 0; s < 8; ++s) {
            int nl = s * 4 + 2 * hi;
            v2f a  = { xt[col * 36 + nl], xt[col * 36 + nl + 1] };
            v2f bb = { tg[nl * 18 + col], tg[(nl + 1) * 18 + col] };
            acc = __builtin_amdgcn_wmma_f32_16x16x4_f32(
                      false, a, false, bb, (short)0, acc, false, false);
        }
    }
    // ---- store raw-sum tile ----
    float* Pb = P + ((size_t)b * CC + ct * 16) * JP + jt * 16;
    #pragma unroll
    for (int r = 0; r < 8; ++r) {
        int row = r + 8 * hi;
        Pb[(size_t)row * JP + col] = acc[r];
    }
}

// ---------------------------------------------------------------------------
// Kernel 3: fold complex weights + 1/(N*M) into interleaved coefficients.
//   k>0:  A_k = [Pc(wr+ + wr-) + Ps(wi+ - wi-)] / (N*M)
//         B_k = [Pc(wi- - wi+) + Ps(wr- + wr+)] / (N*M)
//   k=0:  A_0 = (Pc wr0 + Ps wi0) / (N*M), B_0 = 0
//   k>128 (pad): zeros
// ---------------------------------------------------------------------------
__global__ __launch_bounds__(256)
void coef_kernel(const float* __restrict__ P, const float* __restrict__ wr,
                 const float* __restrict__ wi, float* __restrict__ Coef) {
    int idx = blockIdx.x * blockDim.x + threadIdx.x;     // over B*C*(JP/2)
    if (idx >= BN * CC * (JP / 2)) return;
    int kk = idx % (JP / 2);
    int c  = (idx / (JP / 2)) % CC;
    int b  = idx / (CC * (JP / 2));

    float A = 0.f, Bc = 0.f;
    const float inv = 1.f / ((float)NN * (float)MM);
    if (kk <= KK) {
        size_t base = ((size_t)b * CC + c) * JP;
        float Pc  = P[base + 2 * kk];
        float Ps  = P[base + 2 * kk + 1];
        float wrp = wr[c * MM + KK + kk];
        float wrm = wr[c * MM + KK - kk];
        float wip = wi[c * MM + KK + kk];
        float wim = wi[c * MM + KK - kk];
        if (kk == 0) {
            A  = (Pc * wrp + Ps * wip) * inv;
            Bc = 0.f;
        } else {
            A  = (Pc * (wrp + wrm) + Ps * (wip - wim)) * inv;
            Bc = (Pc * (wim - wip) + Ps * (wrm + wrp)) * inv;
        }
    }
    size_t ob = ((size_t)b * CC + c) * JP;
    Coef[ob + 2 * kk]     = A;
    Coef[ob + 2 * kk + 1] = Bc;
}

// ---------------------------------------------------------------------------
// Kernel 4 (inverse NUFFT GEMM): y[b][c][n] = sum_k A_k cos(k z_n) + B_k sin
// Block = 8 waves, covers one (b, c-tile, 128-n chunk). Coef tile staged in
// LDS once; trig streamed in 16-j slabs with register-carried recurrence.
// grid = (nb=128, ct=8, b=8), block = 256
// ---------------------------------------------------------------------------
__global__ __launch_bounds__(256)
void inv_kernel(const float* __restrict__ Coef, const float* __restrict__ z,
                const float* __restrict__ bias, float* __restrict__ out) {
    const int nb = blockIdx.x, ct = blockIdx.y, b = blockIdx.z;
    const int t    = threadIdx.x;
    const int lane = t & 31, wave = t >> 5;
    const int col  = lane & 15, hi = lane >> 4;
    const int nbase = nb * 128;

    __shared__ float cf[16 * 276];   // coef [c][j], stride 276 (conflict-free, 16B-mult)
    __shared__ float tg[128 * 18];   // trig [n][j-local], stride 18

    // ---- stage coefficient tile 16 x 272 ----
    const float* Cb = Coef + ((size_t)b * CC + ct * 16) * JP;
    for (int idx = t; idx < 16 * JP; idx += 256) {
        int r = idx / JP, j = idx % JP;
        cf[r * 276 + j] = Cb[(size_t)r * JP + j];
    }
    // ---- trig state: thread t<128 owns n = nbase + t ----
    float c1 = 0.f, s1 = 0.f, ck = 1.f, sk = 0.f;
    if (t < 128) {
        float zz = z[(size_t)b * NN + nbase + t];
        sincosf(zz, &s1, &c1);
    }

    v8f acc = {};
    for (int slab = 0; slab < JT; ++slab) {
        __syncthreads();                 // also covers cf staging on slab 0
        if (t < 128) {
            #pragma unroll
            for (int i = 0; i < 8; ++i) {
                tg[t * 18 + 2 * i]     = ck;
                tg[t * 18 + 2 * i + 1] = sk;
                float cn = ck * c1 - sk * s1;
                sk = sk * c1 + ck * s1;
                ck = cn;
            }
        }
        __syncthreads();
        #pragma unroll
        for (int s = 0; s < 4; ++s) {
            int jl = s * 4 + 2 * hi;
            int jj = slab * 16 + jl;
            v2f a  = { cf[col * 276 + jj], cf[col * 276 + jj + 1] };
            int nl = wave * 16 + col;
            v2f bb = { tg[nl * 18 + jl], tg[nl * 18 + jl + 1] };
            acc = __builtin_amdgcn_wmma_f32_16x16x4_f32(
                      false, a, false, bb, (short)0, acc, false, false);
        }
    }
    // ---- epilogue: + bias, store coalesced along n ----
    const int nout = nbase + wave * 16 + col;
    float* ob = out + ((size_t)b * CC + ct * 16) * NN;
    #pragma unroll
    for (int r = 0; r < 8; ++r) {
        int row = r + 8 * hi;
        ob[(size_t)row * NN + nout] = acc[r] + bias[ct * 16 + row];
    }
}

// ---------------------------------------------------------------------------
extern "C" void kernel_launch(void* const* d_in, const int* in_sizes, int n_in,
                              void* d_out, int out_size, void* d_ws, size_t ws_size,
                              hipStream_t stream) {
    const float* x_feat = (const float*)d_in[0];
    const float* z      = (const float*)d_in[1];
    const float* temb   = (const float*)d_in[2];
    const float* wr     = (const float*)d_in[3];
    const float* wi     = (const float*)d_in[4];
    const float* bias   = (const float*)d_in[5];
    const float* tpw    = (const float*)d_in[6];
    const float* tpb    = (const float*)d_in[7];
    float* out = (float*)d_out;

    float* ws    = (float*)d_ws;
    float* shift = ws;                               // BN*CC
    float* P     = ws + 1024;                        // BN*CC*JP
    float* Coef  = P + (size_t)BN * CC * JP;         // BN*CC*JP

    shift_kernel<<<BN, CC, 0, stream>>>(temb, tpw, tpb, shift);

    dim3 g1(CC / 16, JT, BN);
    fwd_kernel<<<g1, 32, 0, stream>>>(x_feat, z, shift, P);

    int tot = BN * CC * (JP / 2);
    coef_kernel<<<(tot + 255) / 256, 256, 0, stream>>>(P, wr, wi, Coef);

    dim3 g2(NN / 128, CC / 16, BN);
    inv_kernel<<<g2, 256, 0, stream>>>(Coef, z, bias, out);
}